// LocalAttention_21500606284364
// MI455X (gfx1250) — compile-verified
//
#include <hip/hip_runtime.h>

typedef __attribute__((ext_vector_type(2))) float v2f;
typedef __attribute__((ext_vector_type(4))) float v4f;
typedef __attribute__((ext_vector_type(8))) float v8f;

#define HALFWIN 2   // WINDOW // 2, WINDOW = 5

// ---------------------------------------------------------------------------
// Kernel 1: predicted window per batch: center = floor(sigmoid(dh.Wp + bp) * S)
// ---------------------------------------------------------------------------
__global__ void win_kernel(const float* __restrict__ dh, const float* __restrict__ Wp,
                           const float* __restrict__ bp, int B, int S, int H,
                           int* __restrict__ wleft, int* __restrict__ wright) {
    int b = blockIdx.x * blockDim.x + threadIdx.x;
    if (b >= B) return;
    float acc = 0.f;
    for (int k = 0; k < H; ++k) acc += dh[b * H + k] * Wp[k];
    acc += bp[0];
    float pos = (1.0f / (1.0f + expf(-acc))) * (float)S;
    int center = (int)floorf(pos);
    wleft[b]  = max(0, center - HALFWIN);
    wright[b] = min(S, center + HALFWIN + 1);
}

// ---------------------------------------------------------------------------
// Kernel 2: q = dh @ Wa  via V_WMMA_F32_16X16X4_F32 (fp32 matrix path)
// One wave per 16x16 output tile; K-loop over H in steps of 4.
//   A (16x4 f32): lanes 0-15 hold M=lane {K0,K1}; lanes 16-31 hold {K2,K3}
//   B (4x16 f32): lanes 0-15 hold N=lane {K0,K1}; lanes 16-31 hold {K2,K3}
//   C/D: VGPR j -> M = j + 8*(lane>=16), N = lane&15
// ---------------------------------------------------------------------------
__global__ void qgemm_wmma(const float* __restrict__ dh, const float* __restrict__ Wa,
                           float* __restrict__ q, int H) {
    int ntilesN = H >> 4;
    int tile = blockIdx.x;
    int tn = tile % ntilesN;
    int tm = tile / ntilesN;
    int m0 = tm << 4, n0 = tn << 4;
    int lane = threadIdx.x & 31;
    int half = lane >> 4;        // 0: lanes 0-15, 1: lanes 16-31
    int l    = lane & 15;

    v8f c = {};
    for (int k0 = 0; k0 < H; k0 += 4) {
        int ka = k0 + half * 2;
        v2f a, bm;
        a.x  = dh[(size_t)(m0 + l) * H + ka];
        a.y  = dh[(size_t)(m0 + l) * H + ka + 1];
        bm.x = Wa[(size_t)ka       * H + n0 + l];
        bm.y = Wa[(size_t)(ka + 1) * H + n0 + l];
        c = __builtin_amdgcn_wmma_f32_16x16x4_f32(false, a, false, bm,
                                                  (short)0, c, false, false);
    }
#pragma unroll
    for (int j = 0; j < 8; ++j) {
        int m = m0 + j + half * 8;
        q[(size_t)m * H + n0 + l] = c[j];
    }
}

// ---------------------------------------------------------------------------
// Kernel 3: partial row-sums of enc over s in [chunk) ∩ [0, L).
// b128 non-temporal streaming loads (512 B / wave / instruction), thread = 4
// consecutive columns.  Deterministic two-stage reduction (no float atomics).
// ---------------------------------------------------------------------------
__global__ void partial_sum_kernel(const v4f* __restrict__ enc4, const int* __restrict__ len,
                                   v4f* __restrict__ part4, int S, int H4,
                                   int rowsPerBlock, int nChunks) {
    int b = blockIdx.y;
    int c = blockIdx.x;
    int L = len[b];
    int s0 = c * rowsPerBlock;
    int s1 = min(s0 + rowsPerBlock, L);
    int h  = threadIdx.x;               // blockDim.x == H4 == H/4
    const v4f* base = enc4 + (size_t)b * S * H4;
    v4f acc = {0.f, 0.f, 0.f, 0.f};
#pragma unroll 4
    for (int s = s0; s < s1; ++s) {
        v4f v = __builtin_nontemporal_load(&base[(size_t)s * H4 + h]);
        acc += v;
    }
    part4[((size_t)b * nChunks + c) * H4 + h] = acc;
}

__global__ void reduce_kernel(const v4f* __restrict__ part4, v4f* __restrict__ SL4,
                              int H4, int nChunks) {
    int b = blockIdx.y;
    int h = blockIdx.x * blockDim.x + threadIdx.x;
    if (h >= H4) return;
    v4f acc = {0.f, 0.f, 0.f, 0.f};
    for (int c = 0; c < nChunks; ++c)
        acc += part4[((size_t)b * nChunks + c) * H4 + h];
    SL4[b * H4 + h] = acc;
}

// ---------------------------------------------------------------------------
// Kernel 4: per-batch finalize.  Window scores (<=5 dots of H), softmax scalars,
// context = w0*SL + sum_{window,valid} (w_s - w0)*enc[s].
// ---------------------------------------------------------------------------
__global__ void finalize_kernel(const float* __restrict__ enc, const float* __restrict__ q,
                                const float* __restrict__ SL, const int* __restrict__ len,
                                const int* __restrict__ wleft, const int* __restrict__ wright,
                                float* __restrict__ ctx_out, float* __restrict__ w0_out,
                                float* __restrict__ wwin_out, int S, int H) {
    int b = blockIdx.x;
    int h = threadIdx.x;                // blockDim.x == H == 512
    int L = len[b];
    int left = wleft[b], right = wright[b];
    int wlo  = min(left, L);
    int whi  = min(right, L);
    int nwin = max(0, whi - wlo);       // # window positions with a real score

    __shared__ float red[512];
    __shared__ float s_scores[8];
    __shared__ float s_w0;
    __shared__ float s_w[5];

    const float* base = enc + (size_t)b * S * H;
    float qh = q[b * H + h];

    float part[5];
#pragma unroll
    for (int i = 0; i < 5; ++i) part[i] = 0.f;
    for (int i = 0; i < nwin; ++i)
        part[i] = qh * base[(size_t)(wlo + i) * H + h];

    for (int i = 0; i < 5; ++i) {
        red[h] = part[i];
        __syncthreads();
        for (int off = H >> 1; off > 0; off >>= 1) {
            if (h < off) red[h] += red[h + off];
            __syncthreads();
        }
        if (h == 0) s_scores[i] = red[0];
        __syncthreads();
    }

    if (h == 0) {
        int nOut = L - nwin;            // valid positions outside window (score 0)
        float m = (nOut > 0) ? 0.f : -3.0e38f;
        for (int i = 0; i < nwin; ++i) m = fmaxf(m, s_scores[i]);
        if (nwin == 0) m = 0.f;
        float Z = (float)nOut * expf(0.f - m);
        float e[5];
        for (int i = 0; i < nwin; ++i) { e[i] = expf(s_scores[i] - m); Z += e[i]; }
        float inv = (Z > 0.f) ? (1.f / Z) : 0.f;
        float w0  = expf(0.f - m) * inv;
        s_w0 = w0;
        w0_out[b] = w0;
        for (int i = 0; i < 5; ++i) {
            float wv = (i < nwin) ? e[i] * inv : 0.f;
            s_w[i] = wv;
            wwin_out[b * 5 + i] = wv;
        }
    }
    __syncthreads();

    float w0 = s_w0;
    float ctxv = w0 * SL[b * H + h];
    for (int i = 0; i < nwin; ++i)
        ctxv += (s_w[i] - w0) * base[(size_t)(wlo + i) * H + h];
    ctx_out[b * H + h] = ctxv;
}

// ---------------------------------------------------------------------------
// Kernel 5: materialize attention_weights [B, S]
// ---------------------------------------------------------------------------
__global__ void weights_kernel(const int* __restrict__ len, const int* __restrict__ wleft,
                               const int* __restrict__ wright, const float* __restrict__ w0,
                               const float* __restrict__ wwin, float* __restrict__ out,
                               int S, int total) {
    int idx = blockIdx.x * blockDim.x + threadIdx.x;
    if (idx >= total) return;
    int b = idx / S;
    int s = idx - b * S;
    int L = len[b];
    float wv;
    if (s >= L)                              wv = 0.f;
    else if (s >= wleft[b] && s < wright[b]) wv = wwin[b * 5 + (s - wleft[b])];
    else                                     wv = w0[b];
    out[idx] = wv;
}

// ---------------------------------------------------------------------------
extern "C" void kernel_launch(void* const* d_in, const int* in_sizes, int n_in,
                              void* d_out, int out_size, void* d_ws, size_t ws_size,
                              hipStream_t stream) {
    const float* dh  = (const float*)d_in[0];   // [B, H]
    const float* enc = (const float*)d_in[1];   // [B, S, H]
    const float* Wp  = (const float*)d_in[2];   // [1, H]
    const float* bp  = (const float*)d_in[3];   // [1]
    const float* Wa  = (const float*)d_in[4];   // [H, H]
    const int*   len = (const int*)d_in[5];     // [B]

    int H = in_sizes[2];                // 512
    int B = in_sizes[0] / H;            // 32
    int S = in_sizes[1] / (B * H);      // 4096
    int H4 = H / 4;

    const int rowsPerBlock = 128;
    int nChunks = (S + rowsPerBlock - 1) / rowsPerBlock;   // 32 -> 1024 blocks

    // workspace layout (floats then ints)
    float* q     = (float*)d_ws;                          // B*H
    float* SL    = q    + (size_t)B * H;                  // B*H
    float* w0    = SL   + (size_t)B * H;                  // B
    float* wwin  = w0   + B;                              // B*5
    float* part  = wwin + (size_t)B * 5 + 2;              // B*nChunks*H (16B-align pad)
    int*   wleft = (int*)(part + (size_t)B * nChunks * H);// B
    int*   wrt   = wleft + B;                             // B

    float* ctx_out  = (float*)d_out;                      // [B, H]
    float* attw_out = ctx_out + (size_t)B * H;            // [B, S]

    win_kernel<<<(B + 31) / 32, 32, 0, stream>>>(dh, Wp, bp, B, S, H, wleft, wrt);

    int tiles = (B / 16) * (H / 16);                      // 64 waves
    qgemm_wmma<<<tiles, 32, 0, stream>>>(dh, Wa, q, H);

    dim3 gsum(nChunks, B);
    partial_sum_kernel<<<gsum, H4, 0, stream>>>((const v4f*)enc, len, (v4f*)part,
                                                S, H4, rowsPerBlock, nChunks);

    dim3 gred((H4 + 127) / 128, B);
    reduce_kernel<<<gred, 128, 0, stream>>>((const v4f*)part, (v4f*)SL, H4, nChunks);

    finalize_kernel<<<B, H, 0, stream>>>(enc, q, SL, len, wleft, wrt,
                                         ctx_out, w0, wwin, S, H);

    int total = B * S;
    weights_kernel<<<(total + 255) / 256, 256, 0, stream>>>(len, wleft, wrt, w0, wwin,
                                                            attw_out, S, total);
}